// EarthquakeGraphSAGE_18949395710312
// MI455X (gfx1250) — compile-verified
//
#include <hip/hip_runtime.h>

#define N_NODES 100000
#define N_EDGES 1600000
#define F_IN 16
#define H 64
#define N_LAYERS 2
#define LN_EPS 1e-5f
#define NTILES (N_NODES / 16)   // 6250, exact
#define WAVES_PER_BLOCK 8

typedef __attribute__((ext_vector_type(2))) float v2f;
typedef __attribute__((ext_vector_type(8))) float v8f;

// D = A(16x4,f32) * B(4x16,f32) + C(16x16,f32)  -- full-precision CDNA5 WMMA
__device__ __forceinline__ v8f wmma4(v2f a, v2f b, v8f c) {
  return __builtin_amdgcn_wmma_f32_16x16x4_f32(
      /*neg_a=*/false, a, /*neg_b=*/false, b,
      /*c_mod=*/(short)0, c, /*reuse_a=*/false, /*reuse_b=*/false);
}

// Hardware f32 atomic add (global_atomic_add_f32), not a CAS loop.
__device__ __forceinline__ void atom_add_f32(float* p, float v) {
  unsafeAtomicAdd(p, v);
}

// ---------------------------------------------------------------------------
// degree / inverse degree
// ---------------------------------------------------------------------------
__global__ __launch_bounds__(256) void deg_kernel(const int* __restrict__ dst,
                                                  float* __restrict__ deg) {
  int e = blockIdx.x * 256 + threadIdx.x;
  if (e < N_EDGES) atom_add_f32(&deg[dst[e]], 1.0f);
}

__global__ __launch_bounds__(256) void invdeg_kernel(float* __restrict__ deg) {
  int i = blockIdx.x * 256 + threadIdx.x;
  if (i < N_NODES) {
    float d = deg[i];
    deg[i] = (d > 0.0f) ? (1.0f / d) : 0.0f;
  }
}

// ---------------------------------------------------------------------------
// input projection: h = relu(x @ Wp^T + bp), x:[N,16], Wp:[64,16]
// one wave per 16-node tile; K=16 -> 4 chained f32 WMMAs per 16x16 out tile
// ---------------------------------------------------------------------------
__global__ __launch_bounds__(256) void proj_kernel(const float* __restrict__ x,
                                                   const float* __restrict__ Wp,
                                                   const float* __restrict__ bp,
                                                   float* __restrict__ h) {
  int wave = (blockIdx.x * 256 + (int)threadIdx.x) >> 5;
  int lane = threadIdx.x & 31;
  if (wave >= NTILES) return;
  int row0 = wave * 16;
  int half = lane >> 4;
  int m    = lane & 15;

  v2f a[4];
#pragma unroll
  for (int ks = 0; ks < 4; ++ks) {
    const float* rp = x + (row0 + m) * F_IN + ks * 4 + half * 2;
    a[ks] = (v2f){rp[0], rp[1]};
  }

#pragma unroll
  for (int ng = 0; ng < 4; ++ng) {
    int n0 = ng * 16;
    int n  = n0 + m;
    v8f c = {};
#pragma unroll
    for (int ks = 0; ks < 4; ++ks) {
      const float* wp = Wp + n * F_IN + ks * 4 + half * 2;   // B[k][n] = Wp[n][k]
      v2f b = (v2f){wp[0], wp[1]};
      c = wmma4(a[ks], b, c);
    }
    float bias = bp[n];
#pragma unroll
    for (int r = 0; r < 8; ++r) {
      int row = row0 + half * 8 + r;
      float v = c[r] + bias;
      h[row * H + n] = v > 0.0f ? v : 0.0f;
    }
  }
}

// ---------------------------------------------------------------------------
// edge scatter: agg[dst] += h[src]  (L2-resident tables, float4 + f32 atomics)
// 16 threads per edge, each moves one float4
// ---------------------------------------------------------------------------
__global__ __launch_bounds__(256) void scatter_kernel(const int* __restrict__ src,
                                                      const int* __restrict__ dst,
                                                      const float* __restrict__ h,
                                                      float* __restrict__ agg) {
  int gid = blockIdx.x * 256 + threadIdx.x;          // N_EDGES*16 = 25.6M < 2^31
  if (gid >= N_EDGES * 16) return;
  int e = gid >> 4;
  int f = (gid & 15) * 4;
  int s = src[e];
  int d = dst[e];
  const float4 v = *(const float4*)(h + s * H + f);
  float* ap = agg + d * H + f;
  atom_add_f32(ap + 0, v.x);
  atom_add_f32(ap + 1, v.y);
  atom_add_f32(ap + 2, v.z);
  atom_add_f32(ap + 3, v.w);
}

// ---------------------------------------------------------------------------
// SAGE layer: out = (agg*invdeg) @ Wl^T + bl + h @ Wr^T ; LN ; relu + identity
// one wave per 16-node tile; K=64 -> 16 chained WMMAs per GEMM per 16x16 tile
// ---------------------------------------------------------------------------
__global__ __launch_bounds__(256) void sage_kernel(const float* __restrict__ h,
                                                   const float* __restrict__ agg,
                                                   const float* __restrict__ invdeg,
                                                   const float* __restrict__ Wl,
                                                   const float* __restrict__ bl,
                                                   const float* __restrict__ Wr,
                                                   const float* __restrict__ gamma,
                                                   const float* __restrict__ beta,
                                                   float* __restrict__ hout) {
  __shared__ float lds_all[WAVES_PER_BLOCK][16 * H];   // 32 KB
  int wib  = threadIdx.x >> 5;
  int lane = threadIdx.x & 31;
  int wave = blockIdx.x * WAVES_PER_BLOCK + wib;
  bool active = (wave < NTILES);
  float* lds = lds_all[wib];
  int row0 = wave * 16;
  int half = lane >> 4;
  int m    = lane & 15;

  if (active) {
    float idg = invdeg[row0 + m];
    v2f aA[16], aH[16];
#pragma unroll
    for (int ks = 0; ks < 16; ++ks) {
      const float* pa = agg + (row0 + m) * H + ks * 4 + half * 2;
      aA[ks] = (v2f){pa[0] * idg, pa[1] * idg};
      const float* ph = h + (row0 + m) * H + ks * 4 + half * 2;
      aH[ks] = (v2f){ph[0], ph[1]};
    }
#pragma unroll
    for (int ng = 0; ng < 4; ++ng) {
      int n0 = ng * 16;
      int n  = n0 + m;
      v8f c = {};
#pragma unroll
      for (int ks = 0; ks < 16; ++ks) {
        const float* wp = Wl + n * H + ks * 4 + half * 2;
        v2f b = (v2f){wp[0], wp[1]};
        c = wmma4(aA[ks], b, c);
      }
#pragma unroll
      for (int ks = 0; ks < 16; ++ks) {
        const float* wp = Wr + n * H + ks * 4 + half * 2;
        v2f b = (v2f){wp[0], wp[1]};
        c = wmma4(aH[ks], b, c);
      }
      float bias = bl[n];
#pragma unroll
      for (int r = 0; r < 8; ++r)
        lds[(half * 8 + r) * H + n] = c[r] + bias;
    }
  }
  __syncthreads();

  // row-wise LayerNorm + relu + residual; lanes 0..15 each own one row
  if (active && lane < 16) {
    const float* rp = lds + lane * H;
    float mu = 0.0f;
#pragma unroll
    for (int j = 0; j < H; ++j) mu += rp[j];
    mu *= (1.0f / H);
    float var = 0.0f;
#pragma unroll
    for (int j = 0; j < H; ++j) {
      float d = rp[j] - mu;
      var += d * d;
    }
    var *= (1.0f / H);
    float rstd = rsqrtf(var + LN_EPS);
    int grow = row0 + lane;
    for (int j = 0; j < H; ++j) {
      float v = (rp[j] - mu) * rstd * gamma[j] + beta[j];
      v = v > 0.0f ? v : 0.0f;
      hout[grow * H + j] = v + h[grow * H + j];   // residual = layer input
    }
  }
}

// ---------------------------------------------------------------------------
// MLP head: t = relu(h @ W1^T + b1) [N,32]; y = t @ W2^T + b2 [N]
// ---------------------------------------------------------------------------
__global__ __launch_bounds__(256) void mlp_kernel(const float* __restrict__ h,
                                                  const float* __restrict__ W1,
                                                  const float* __restrict__ b1,
                                                  const float* __restrict__ W2,
                                                  const float* __restrict__ b2,
                                                  float* __restrict__ y) {
  __shared__ float lds_all[WAVES_PER_BLOCK][16 * 32];  // 16 KB
  int wib  = threadIdx.x >> 5;
  int lane = threadIdx.x & 31;
  int wave = blockIdx.x * WAVES_PER_BLOCK + wib;
  bool active = (wave < NTILES);
  float* lds = lds_all[wib];
  int row0 = wave * 16;
  int half = lane >> 4;
  int m    = lane & 15;

  if (active) {
    v2f aH[16];
#pragma unroll
    for (int ks = 0; ks < 16; ++ks) {
      const float* ph = h + (row0 + m) * H + ks * 4 + half * 2;
      aH[ks] = (v2f){ph[0], ph[1]};
    }
#pragma unroll
    for (int ng = 0; ng < 2; ++ng) {
      int n = ng * 16 + m;
      v8f c = {};
#pragma unroll
      for (int ks = 0; ks < 16; ++ks) {
        const float* wp = W1 + n * H + ks * 4 + half * 2;
        v2f b = (v2f){wp[0], wp[1]};
        c = wmma4(aH[ks], b, c);
      }
      float bias = b1[n];
#pragma unroll
      for (int r = 0; r < 8; ++r) {
        float v = c[r] + bias;
        lds[(half * 8 + r) * 32 + n] = v > 0.0f ? v : 0.0f;
      }
    }
  }
  __syncthreads();

  if (active && lane < 16) {
    float acc = b2[0];
    const float* rp = lds + lane * 32;
#pragma unroll
    for (int j = 0; j < 32; ++j) acc += rp[j] * W2[j];
    y[row0 + lane] = acc;
  }
}

// ---------------------------------------------------------------------------
extern "C" void kernel_launch(void* const* d_in, const int* in_sizes, int n_in,
                              void* d_out, int out_size, void* d_ws, size_t ws_size,
                              hipStream_t stream) {
  const float* x     = (const float*)d_in[0];
  const int*   eidx  = (const int*)d_in[1];     // [2, N_EDGES]
  const float* Wp    = (const float*)d_in[2];
  const float* bp    = (const float*)d_in[3];
  const float* Wl    = (const float*)d_in[4];   // [2,64,64]
  const float* bl    = (const float*)d_in[5];   // [2,64]
  const float* Wr    = (const float*)d_in[6];   // [2,64,64]
  const float* gamma = (const float*)d_in[7];   // [2,64]
  const float* beta  = (const float*)d_in[8];   // [2,64]
  const float* W1    = (const float*)d_in[9];   // [32,64]
  const float* b1    = (const float*)d_in[10];  // [32]
  const float* W2    = (const float*)d_in[11];  // [1,32]
  const float* b2    = (const float*)d_in[12];  // [1]
  float*       y     = (float*)d_out;

  const int* src = eidx;
  const int* dst = eidx + N_EDGES;

  // workspace carve-out: hA | hB | agg | invdeg  (~77.2 MB of floats)
  float* hA   = (float*)d_ws;
  float* hB   = hA + (size_t)N_NODES * H;
  float* agg  = hB + (size_t)N_NODES * H;
  float* ideg = agg + (size_t)N_NODES * H;

  const int tile_blocks = (NTILES + WAVES_PER_BLOCK - 1) / WAVES_PER_BLOCK;  // 782

  // degree -> inverse degree (ideg doubles as deg accumulator)
  hipMemsetAsync(ideg, 0, (size_t)N_NODES * sizeof(float), stream);
  deg_kernel<<<(N_EDGES + 255) / 256, 256, 0, stream>>>(dst, ideg);
  invdeg_kernel<<<(N_NODES + 255) / 256, 256, 0, stream>>>(ideg);

  // input projection
  proj_kernel<<<tile_blocks, 256, 0, stream>>>(x, Wp, bp, hA);

  float* hin  = hA;
  float* hout = hB;
  for (int l = 0; l < N_LAYERS; ++l) {
    hipMemsetAsync(agg, 0, (size_t)N_NODES * H * sizeof(float), stream);
    scatter_kernel<<<(N_EDGES * 16) / 256, 256, 0, stream>>>(src, dst, hin, agg);
    sage_kernel<<<tile_blocks, 256, 0, stream>>>(hin, agg, ideg,
                                                 Wl + (size_t)l * H * H, bl + (size_t)l * H,
                                                 Wr + (size_t)l * H * H,
                                                 gamma + (size_t)l * H, beta + (size_t)l * H,
                                                 hout);
    float* t = hin; hin = hout; hout = t;
  }

  mlp_kernel<<<tile_blocks, 256, 0, stream>>>(hin, W1, b1, W2, b2, y);
}